// CognateACTLayer_62594853372108
// MI455X (gfx1250) — compile-verified
//
#include <hip/hip_runtime.h>

// Problem constants (from reference)
#define BDIM 16
#define SDIM 2048
#define DDIM 512
#define NROWS (BDIM * SDIM)               // 32768
#define ROWS_PER_BLOCK 16
#define NBLOCKS (NROWS / ROWS_PER_BLOCK)  // 2048
#define NSTEPS 10
#define THRESH 0.99f

// W streaming geometry: chunks of 16 K-rows; TDM pads 8 DWORDs per 256 DWORDs
// -> effective LDS row stride 528 DWORDs (16-bank shift per K-row, so the two
//    lane halves of a B fragment hit disjoint bank sets). Triple-buffered:
//    the single per-chunk barrier both publishes chunk c and proves the
//    buffer for chunk c+2 (last read at c-1) is free.
#define WROW 528
#define WCHUNK_ROWS 16
#define WCHUNK_DW (WCHUNK_ROWS * WROW)    // 8448 dwords / chunk buffer
#define WCHUNK_BYTES (WCHUNK_DW * 4)      // 33,792 B
#define NCHUNK 32
#define NBUF 3

typedef float v2f __attribute__((ext_vector_type(2)));
typedef float v8f __attribute__((ext_vector_type(8)));
typedef unsigned int u32x4 __attribute__((ext_vector_type(4)));
typedef int i32x4 __attribute__((ext_vector_type(4)));
typedef int i32x8 __attribute__((ext_vector_type(8)));

#if defined(__has_builtin)
#if __has_builtin(__builtin_amdgcn_tensor_load_to_lds)
#define USE_TDM 1
#endif
#endif

#if defined(USE_TDM)
// Issue one TDM load: W rows [chunk*16, chunk*16+16) -> LDS buffer at lds_addr.
// D# per CDNA5 ISA ch.8 (group0: count/lds/global/type, group1: geometry).
__device__ __forceinline__ void tdm_load_chunk(const float* __restrict__ Wl,
                                               int chunk, unsigned lds_addr) {
  unsigned long long ga =
      (unsigned long long)(const void*)(Wl + (size_t)chunk * WCHUNK_ROWS * DDIM);
  u32x4 g0;
  g0[0] = 1u;                                   // count=1 (valid user descriptor)
  g0[1] = lds_addr;                             // LDS byte address of tile start
  g0[2] = (unsigned)ga;                         // global_addr[31:0]
  g0[3] = (unsigned)((ga >> 32) & 0x1FFFFFFu)   // global_addr[56:32]
        | (2u << 30);                           // type = 2 ("image")
  i32x8 g1;
  g1[0] = (int)((2u << 16)      // data_size = 4 bytes
              | (1u << 20)      // pad_enable
              | (7u << 22)      // pad_interval: 256 DWORDs
              | (7u << 25));    // pad_amount:   8 DWORDs
  g1[1] = (int)(512u << 16);    // tensor_dim0 = 512 (bits 79:48, low part)
  g1[2] = (int)(16u << 16);     // tensor_dim1 = 16  (bits 111:80, low part)
  g1[3] = (int)(512u << 16);    // tile_dim0 = 512   (bits 127:112)
  g1[4] = 16;                   // tile_dim1 = 16    (bits 143:128)
  g1[5] = 512;                  // tensor_dim0_stride = 512 (bits 207:160)
  g1[6] = 0;
  g1[7] = 0;
  i32x4 z4 = {0, 0, 0, 0};
#if defined(__clang_major__) && (__clang_major__ >= 23)
  i32x8 z8 = {0, 0, 0, 0, 0, 0, 0, 0};
  __builtin_amdgcn_tensor_load_to_lds(g0, g1, z4, z4, z8, 0);
#else
  __builtin_amdgcn_tensor_load_to_lds(g0, g1, z4, z4, 0);
#endif
}
#endif

// ---------------------------------------------------------------------------
// One block = 16 rows; state tile resident in LDS for all 10 ACT steps.
// layer_out = state @ W_layer via V_WMMA_F32_16X16X4_F32; W streamed by TDM.
// ---------------------------------------------------------------------------
__global__ __launch_bounds__(256) void act_wmma_kernel(
    const float* __restrict__ x,      // (B,S,D)
    const float* __restrict__ Wh,     // (D,1)
    const float* __restrict__ bh,     // (1,)
    const float* __restrict__ Wl,     // (D,D)
    const float* __restrict__ bl,     // (D,)
    float* __restrict__ out,          // (B,S,D)
    float* __restrict__ ws)           // per-block partials: 3 floats / block
{
  // Static LDS (37.4 KB): state tile padded to 516 (A-fragment b64 reads
  // cover all 64 banks exactly once), W_halt, bias, ACT scalars.
  __shared__ float stateS[16][516];
  __shared__ float WhS[DDIM];
  __shared__ float biasS[DDIM];
  __shared__ float pS[16], hpS[16], remS[16], nupS[16];
  // Dynamic LDS (101.4 KB): three W-chunk buffers (TDM destinations).
  extern __shared__ float WcDyn[];

  const int tid    = threadIdx.x;
  const int lane   = tid & 31;
  const int wave   = tid >> 5;        // 0..7 -> owns N columns [64w, 64w+64)
  const int lanelo = lane & 15;
  const int hi     = lane >> 4;

  const long rowBase = (long)blockIdx.x * ROWS_PER_BLOCK;
  // LDS aperture: low 32 bits of the generic address are the LDS byte offset.
  const unsigned ldsBase = (unsigned)(size_t)(void*)WcDyn;
  // Columns >= 256 sit after the mid-row TDM pad (8 DWORDs).
  const int colOff = wave * 64 + ((wave >= 4) ? 8 : 0);
  const int colAbs = wave * 64;       // unpadded column base (bias/out/state)

  // ---- init small LDS ----
  WhS[tid]         = Wh[tid];
  WhS[tid + 256]   = Wh[tid + 256];
  biasS[tid]       = bl[tid];
  biasS[tid + 256] = bl[tid + 256];
  if (tid < 16) { pS[tid] = 0.f; hpS[tid] = 0.f; remS[tid] = 1.f; nupS[tid] = 0.f; }

  // ---- load x tile (16x512) into stateS ----
  {
    const float4* src = (const float4*)(x + rowBase * DDIM);
#pragma unroll
    for (int i = 0; i < 8; ++i) {
      int idx = tid + i * 256;          // 0..2047 float4s
      float4 v = src[idx];
      int r = idx >> 7, c4 = idx & 127;
      *(float4*)&stateS[r][c4 * 4] = v;
    }
  }
  __syncthreads();

  const float bhv = bh[0];
  v8f oacc[4] = {};
  const int r0 = wave * 2, r1 = r0 + 1;  // halting rows owned by this wave

  for (int step = 0; step < NSTEPS; ++step) {
#if defined(USE_TDM)
    if (wave == 0) {                     // prologue: two chunks in flight
      tdm_load_chunk(Wl, 0, ldsBase);
      tdm_load_chunk(Wl, 1, ldsBase + WCHUNK_BYTES);
    }
#endif

    // -------- halting probe: p_lin = state . W_halt for rows r0,r1 --------
    float s0 = 0.f, s1 = 0.f;
    for (int k = lane; k < DDIM; k += 32) {
      float w = WhS[k];
      s0 += stateS[r0][k] * w;
      s1 += stateS[r1][k] * w;
    }
#pragma unroll
    for (int m = 16; m > 0; m >>= 1) {
      s0 += __shfl_xor(s0, m, 32);
      s1 += __shfl_xor(s1, m, 32);
    }
    if (lane == 0) {
#pragma unroll
      for (int rr = 0; rr < 2; ++rr) {
        int r = r0 + rr;
        float sv    = rr ? s1 : s0;
        float p     = 1.f / (1.f + __expf(-(sv + bhv)));
        float hpo   = hpS[r];
        float still = (hpo < THRESH) ? 1.f : 0.f;
        float ps    = p * still;
        float peff  = (hpo + ps >= THRESH) ? remS[r] : ps;
        pS[r]   = peff;
        hpS[r]  = hpo + peff;
        remS[r] = remS[r] - peff;
        nupS[r] = nupS[r] + still;
      }
    }

    // -------- layer_out = state @ W_layer (WMMA f32 16x16x4) --------
    v8f cur[4] = {};
    int ib = 0;                          // rotating buffer index of chunk c

    for (int c = 0; c < NCHUNK; ++c) {
#if defined(USE_TDM)
      if (wave == 0) {                   // in-order TDM: oldest chunk landed
        if (c == NCHUNK - 1) __builtin_amdgcn_s_wait_tensorcnt(0);
        else                 __builtin_amdgcn_s_wait_tensorcnt(1);
      }
      __syncthreads();                   // publish chunk c; frees buf of c-1
      if (wave == 0 && c + 2 < NCHUNK) { // refill buffer last read at c-1
        int ii = ib + 2; if (ii >= NBUF) ii -= NBUF;
        tdm_load_chunk(Wl, c + 2, ldsBase + (unsigned)ii * WCHUNK_BYTES);
      }
#else
      // Fallback: cooperative global->LDS staging, identical geometry.
      {
        float* __restrict__ wbw = WcDyn + ib * WCHUNK_DW;
        const float4* src = (const float4*)(Wl + (size_t)c * WCHUNK_ROWS * DDIM);
#pragma unroll
        for (int i = 0; i < 8; ++i) {
          int idx = tid + i * 256;       // 0..2047 float4s
          float4 v = src[idx];
          int r = idx >> 7, c4 = idx & 127;
          int col = c4 * 4;
          *(float4*)&wbw[r * WROW + col + ((col >= 256) ? 8 : 0)] = v;
        }
      }
      __syncthreads();                   // staged chunk visible to all waves
#endif
      const float* __restrict__ wb = WcDyn + ib * WCHUNK_DW;

#pragma unroll
      for (int kk = 0; kk < 4; ++kk) {   // four K=4 blocks per chunk
        const int kglob = c * 16 + kk * 4 + hi * 2; // A: lanes0-15 K=0,1; 16-31 K=2,3
        const v2f a = *(const v2f*)&stateS[lanelo][kglob];
        const int krow = kk * 4 + hi * 2;
#pragma unroll
        for (int t = 0; t < 4; ++t) {
          const int ci = colOff + t * 16 + lanelo;
          v2f b;
          b[0] = wb[krow * WROW + ci];
          b[1] = wb[(krow + 1) * WROW + ci];
          cur[t] = __builtin_amdgcn_wmma_f32_16x16x4_f32(
              /*neg_a=*/false, a, /*neg_b=*/false, b,
              /*c_mod=*/(short)0, cur[t], /*reuse_a=*/false, /*reuse_b=*/false);
        }
      }

      ib = (ib + 1 == NBUF) ? 0 : ib + 1;
    }

    // -------- bias, out accumulation: out += p[row] * layer_out --------
    float pr[8];
#pragma unroll
    for (int i = 0; i < 8; ++i) pr[i] = pS[i + 8 * hi];  // row = i + 8*hi
#pragma unroll
    for (int t = 0; t < 4; ++t) {
      const float bb = biasS[colAbs + t * 16 + lanelo];
#pragma unroll
      for (int i = 0; i < 8; ++i) {
        float lo = cur[t][i] + bb;
        cur[t][i] = lo;
        oacc[t][i] += pr[i] * lo;
      }
    }

    // -------- state <- layer_out --------
    __syncthreads();                     // all reads of old state done
#pragma unroll
    for (int t = 0; t < 4; ++t) {
      const int col = colAbs + t * 16 + lanelo;
#pragma unroll
      for (int i = 0; i < 8; ++i) stateS[i + 8 * hi][col] = cur[t][i];
    }
    __syncthreads();                     // new state visible
  }

  // -------- write out tile --------
#pragma unroll
  for (int t = 0; t < 4; ++t) {
    const int col = colAbs + t * 16 + lanelo;
#pragma unroll
    for (int i = 0; i < 8; ++i) {
      const long row = rowBase + i + 8 * hi;
      out[row * DDIM + col] = oacc[t][i];
    }
  }

  // -------- per-block scalar partials --------
  if (tid == 0) {
    float sp = 0.f, sn = 0.f, sh = 0.f;
    for (int r = 0; r < 16; ++r) {
      sp += nupS[r] + remS[r];
      sn += nupS[r];
      sh += hpS[r];
    }
    ws[blockIdx.x * 3 + 0] = sp;
    ws[blockIdx.x * 3 + 1] = sn;
    ws[blockIdx.x * 3 + 2] = sh;
  }
}

// ---------------------------------------------------------------------------
// Fixed-order reduction of per-block partials -> 3 scalar outputs.
// ---------------------------------------------------------------------------
__global__ __launch_bounds__(256) void act_reduce_kernel(
    const float* __restrict__ ws, float* __restrict__ scal)
{
  __shared__ float acc[3][256];
  const int t = threadIdx.x;
  float a0 = 0.f, a1 = 0.f, a2 = 0.f;
  for (int i = t; i < NBLOCKS; i += 256) {
    a0 += ws[i * 3 + 0];
    a1 += ws[i * 3 + 1];
    a2 += ws[i * 3 + 2];
  }
  acc[0][t] = a0; acc[1][t] = a1; acc[2][t] = a2;
  __syncthreads();
  for (int s = 128; s > 0; s >>= 1) {
    if (t < s) {
      acc[0][t] += acc[0][t + s];
      acc[1][t] += acc[1][t + s];
      acc[2][t] += acc[2][t + s];
    }
    __syncthreads();
  }
  if (t == 0) {
    const float inv = 1.f / (float)NROWS;
    scal[0] = acc[0][0] * inv;   // ponder_cost
    scal[1] = acc[1][0] * inv;   // avg_steps
    scal[2] = acc[2][0] * inv;   // halting_prob_mean
  }
}

// ---------------------------------------------------------------------------
extern "C" void kernel_launch(void* const* d_in, const int* in_sizes, int n_in,
                              void* d_out, int out_size, void* d_ws, size_t ws_size,
                              hipStream_t stream) {
  const float* x  = (const float*)d_in[0];   // (16,2048,512)
  const float* Wh = (const float*)d_in[1];   // (512,1)
  const float* bh = (const float*)d_in[2];   // (1,)
  const float* Wl = (const float*)d_in[3];   // (512,512)
  const float* bl = (const float*)d_in[4];   // (512,)

  float* out  = (float*)d_out;               // 16*2048*512 floats + 3 scalars
  float* scal = out + (size_t)NROWS * DDIM;
  float* ws   = (float*)d_ws;                // needs NBLOCKS*3 floats (24 KB)

  const size_t dynLds = (size_t)NBUF * WCHUNK_BYTES;  // 101,376 B
  act_wmma_kernel<<<NBLOCKS, 256, dynLds, stream>>>(x, Wh, bh, Wl, bl, out, ws);
  act_reduce_kernel<<<1, 256, 0, stream>>>(ws, scal);
}